// GAT_36094905155903
// MI455X (gfx1250) — compile-verified
//
#include <hip/hip_runtime.h>
#include <math.h>

typedef __attribute__((ext_vector_type(2))) float v2f;
typedef __attribute__((ext_vector_type(8))) float v8f;

#define NEG_SLOPE 0.2f
#define BN_EPS 1e-5f

__device__ __forceinline__ int f2ord(float f) {
    int i = __float_as_int(f);
    return i >= 0 ? i : (i ^ 0x7FFFFFFF);
}
__device__ __forceinline__ float ord2f(int i) {
    return __int_as_float(i >= 0 ? i : (i ^ 0x7FFFFFFF));
}
__device__ __forceinline__ float lrelu(float x) { return x > 0.f ? x : NEG_SLOPE * x; }

// ---------------- fills ----------------
__global__ void fill_f32(float* __restrict__ p, float v, int n) {
    int i = blockIdx.x * blockDim.x + threadIdx.x;
    if (i < n) p[i] = v;
}
__global__ void fill_i32(int* __restrict__ p, int v, int n) {
    int i = blockIdx.x * blockDim.x + threadIdx.x;
    if (i < n) p[i] = v;
}

// ---------------- fp32 WMMA GEMM: C[M,Nc] = A[M,K] * B[K,Nc] ----------------
// One wave per 16x16 output tile; V_WMMA_F32_16X16X4_F32, K stepped by 4.
// A frag (16x4): lanes 0-15 M=0..15, VGPR0 = K (lanes<16) / K+2 (lanes>=16), VGPR1 = +1.
// B frag (4x16): row K striped across lanes 0-15 (VGPR0), K+1 in VGPR1; rows K+2/K+3 in lanes 16-31.
// D: VGPR r holds row (r + 8*(lane>=16)), col = lane&15.
__global__ void gemm_wmma_f32(const float* __restrict__ A, const float* __restrict__ B,
                              float* __restrict__ C, int M, int K, int Nc) {
    const int lane = threadIdx.x;         // 0..31
    const int l15  = lane & 15;
    const int half = lane >> 4;           // 0 or 1
    const int mt   = blockIdx.x * 16;
    const int nt   = (blockIdx.y * blockDim.y + threadIdx.y) * 16;
    if (mt >= M || nt >= Nc) return;      // uniform per wave

    const float* __restrict__ Arow = A + (size_t)(mt + l15) * K;
    v8f acc = {};
    for (int k = 0; k < K; k += 4) {
        const int ka = k + half * 2;
        v2f av;
        av.x = Arow[ka];
        av.y = Arow[ka + 1];
        v2f bv;
        bv.x = B[(size_t)ka * Nc + nt + l15];
        bv.y = B[(size_t)(ka + 1) * Nc + nt + l15];
        acc = __builtin_amdgcn_wmma_f32_16x16x4_f32(false, av, false, bv,
                                                    (short)0, acc, false, false);
    }
#pragma unroll
    for (int r = 0; r < 8; ++r)
        C[(size_t)(mt + half * 8 + r) * Nc + nt + l15] = acc[r];
}

// ---------------- attention logits: al[n,h] = dot(hfeat[n,h,:], a[h,:]) ----------------
__global__ void attn_logits(const float* __restrict__ h, const float* __restrict__ a_s,
                            const float* __restrict__ a_d, float* __restrict__ als,
                            float* __restrict__ ald, int N, int Cc) {
    const int H = 4;
    int idx = blockIdx.x * blockDim.x + threadIdx.x;
    if (idx >= N * H) return;
    int n = idx / H, hd = idx % H;
    const float* hp  = h + (size_t)n * H * Cc + hd * Cc;
    const float* asp = a_s + hd * Cc;
    const float* adp = a_d + hd * Cc;
    float ss = 0.f, sd = 0.f;
    for (int c = 0; c < Cc; c += 4) {
        float4 v  = *(const float4*)(hp + c);
        float4 w1 = *(const float4*)(asp + c);
        float4 w2 = *(const float4*)(adp + c);
        ss += v.x * w1.x + v.y * w1.y + v.z * w1.z + v.w * w1.w;
        sd += v.x * w2.x + v.y * w2.y + v.z * w2.z + v.w * w2.w;
    }
    als[idx] = ss;
    ald[idx] = sd;
}

// ---------------- edge pass A: e = leakyrelu(al_s[src]+al_d[dst]); atomic segment max ----------------
__global__ void edge_logits_max(const int* __restrict__ srcp, const int* __restrict__ dstp,
                                int E, int Etot, const float* __restrict__ als,
                                const float* __restrict__ ald, float* __restrict__ pbuf,
                                int* __restrict__ menc) {
    int e = blockIdx.x * blockDim.x + threadIdx.x;
    if (e >= Etot) return;
    int s_, d_;
    if (e < E) { s_ = srcp[e]; d_ = dstp[e]; } else { s_ = d_ = e - E; }
    float4 a = *(const float4*)(als + (size_t)s_ * 4);
    float4 b = *(const float4*)(ald + (size_t)d_ * 4);
    float4 ev;
    ev.x = lrelu(a.x + b.x); ev.y = lrelu(a.y + b.y);
    ev.z = lrelu(a.z + b.z); ev.w = lrelu(a.w + b.w);
    *(float4*)(pbuf + (size_t)e * 4) = ev;
    int* mb = menc + (size_t)d_ * 4;
    atomicMax(mb + 0, f2ord(ev.x));
    atomicMax(mb + 1, f2ord(ev.y));
    atomicMax(mb + 2, f2ord(ev.z));
    atomicMax(mb + 3, f2ord(ev.w));
}

// ---------------- edge pass B: p = exp(e - m[dst]); atomic segment sum ----------------
__global__ void edge_exp_sum(const int* __restrict__ dstp, int E, int Etot, int N,
                             float* __restrict__ pbuf, const int* __restrict__ menc,
                             float* __restrict__ ssum) {
    int e = blockIdx.x * blockDim.x + threadIdx.x;
    if (e >= Etot) return;
    int d_ = (e < E) ? dstp[e] : (e - E);
    float4 ev = *(const float4*)(pbuf + (size_t)e * 4);
    const int* mb = menc + (size_t)d_ * 4;
    float4 pv;
    pv.x = __expf(ev.x - ord2f(mb[0]));
    pv.y = __expf(ev.y - ord2f(mb[1]));
    pv.z = __expf(ev.z - ord2f(mb[2]));
    pv.w = __expf(ev.w - ord2f(mb[3]));
    *(float4*)(pbuf + (size_t)e * 4) = pv;
    float* sb = ssum + (size_t)d_ * 4;
    atomicAdd(sb + 0, pv.x);
    atomicAdd(sb + 1, pv.y);
    atomicAdd(sb + 2, pv.z);
    atomicAdd(sb + 3, pv.w);
}

// ---------------- edge pass C: agg[dst] += alpha * h[src] ----------------
// thread = (edge, float4 chunk); consecutive threads read contiguous source row -> coalesced
__global__ void edge_aggregate(const int* __restrict__ srcp, const int* __restrict__ dstp,
                               int E, int Etot, const float* __restrict__ h,
                               const float* __restrict__ pbuf, const float* __restrict__ ssum,
                               float* __restrict__ agg, int K, int Cc) {
    int chunks = K >> 2;
    int gid = blockIdx.x * blockDim.x + threadIdx.x;
    if (gid >= Etot * chunks) return;
    int e  = gid / chunks;
    int ch = gid - e * chunks;
    int s_, d_;
    if (e < E) { s_ = srcp[e]; d_ = dstp[e]; } else { s_ = d_ = e - E; }
    int hd = (ch << 2) / Cc;
    float alpha = pbuf[(size_t)e * 4 + hd] / ssum[(size_t)d_ * 4 + hd];
    float4 hv = *(const float4*)(h + (size_t)s_ * K + (ch << 2));
    float* o = agg + (size_t)d_ * K + (ch << 2);
    atomicAdd(o + 0, alpha * hv.x);
    atomicAdd(o + 1, alpha * hv.y);
    atomicAdd(o + 2, alpha * hv.z);
    atomicAdd(o + 3, alpha * hv.w);
}

// ---------------- batchnorm stats: per-channel sum/sumsq of (agg + bias) ----------------
__global__ void bn_stats(const float* __restrict__ agg, const float* __restrict__ bias,
                         float* __restrict__ stats, int N, int K) {
    int t = threadIdx.x;  // channel, blockDim.x == K
    float sum = 0.f, sq = 0.f;
    for (int r = blockIdx.x; r < N; r += gridDim.x) {
        float v = agg[(size_t)r * K + t] + bias[t];
        sum += v;
        sq += v * v;
    }
    atomicAdd(&stats[t], sum);
    atomicAdd(&stats[K + t], sq);
}

__global__ void bn_finalize(float* __restrict__ stats, int N, int K) {
    int t = threadIdx.x;
    if (t >= K) return;
    float mu  = stats[t] / (float)N;
    float var = stats[K + t] / (float)N - mu * mu;
    stats[2 * K + t] = mu;
    stats[3 * K + t] = rsqrtf(var + BN_EPS);
}

// ---------------- fused bias + batchnorm + elu ----------------
__global__ void bn_apply_elu(const float* __restrict__ agg, const float* __restrict__ bias,
                             const float* __restrict__ g, const float* __restrict__ be,
                             const float* __restrict__ stats, float* __restrict__ out,
                             int N, int K) {
    int gid = blockIdx.x * blockDim.x + threadIdx.x;
    if (gid >= N * K) return;
    int c = gid % K;
    float v = agg[gid] + bias[c];
    float y = g[c] * (v - stats[2 * K + c]) * stats[3 * K + c] + be[c];
    out[gid] = y > 0.f ? y : expm1f(y);
}

// ---------------- layer-3 epilogue: mean over 4 heads + bias ----------------
__global__ void head_mean_bias(const float* __restrict__ agg, const float* __restrict__ b3,
                               float* __restrict__ out, int N) {
    const int OUT = 64;
    int gid = blockIdx.x * blockDim.x + threadIdx.x;
    if (gid >= N * OUT) return;
    int n = gid / OUT, c = gid - n * OUT;
    const float* a = agg + (size_t)n * 256;
    out[gid] = 0.25f * (a[c] + a[64 + c] + a[128 + c] + a[192 + c]) + b3[c];
}

// ---------------- host-side layer driver ----------------
static void run_gat_layer(const float* feat, int Fin, const float* W, const float* a_s,
                          const float* a_d, int Cc, const int* srcp, const int* dstp,
                          int E, int N, float* h, float* agg, float* pbuf, float* als,
                          float* ald, float* ssum, int* menc, hipStream_t stream) {
    const int H = 4;
    const int K = H * Cc;
    const int Etot = E + N;

    dim3 gb(32, 4, 1);
    dim3 gg(N / 16, K / 64, 1);
    gemm_wmma_f32<<<gg, gb, 0, stream>>>(feat, W, h, N, Fin, K);

    attn_logits<<<(N * H + 255) / 256, 256, 0, stream>>>(h, a_s, a_d, als, ald, N, Cc);

    fill_i32<<<(N * H + 255) / 256, 256, 0, stream>>>(menc, (int)0x80000000, N * H);
    fill_f32<<<(N * H + 255) / 256, 256, 0, stream>>>(ssum, 0.f, N * H);
    fill_f32<<<((size_t)N * K + 255) / 256, 256, 0, stream>>>(agg, 0.f, N * K);

    edge_logits_max<<<(Etot + 255) / 256, 256, 0, stream>>>(srcp, dstp, E, Etot, als, ald,
                                                            pbuf, menc);
    edge_exp_sum<<<(Etot + 255) / 256, 256, 0, stream>>>(dstp, E, Etot, N, pbuf, menc, ssum);

    long long tc = (long long)Etot * (K / 4);
    edge_aggregate<<<(unsigned)((tc + 255) / 256), 256, 0, stream>>>(srcp, dstp, E, Etot, h,
                                                                     pbuf, ssum, agg, K, Cc);
}

static void run_bn_elu(const float* agg, const float* bias, const float* g, const float* be,
                       float* stats, float* out, int N, int K, hipStream_t stream) {
    fill_f32<<<(4 * K + 255) / 256, 256, 0, stream>>>(stats, 0.f, 4 * K);
    bn_stats<<<512, K, 0, stream>>>(agg, bias, stats, N, K);
    bn_finalize<<<1, K, 0, stream>>>(stats, N, K);
    bn_apply_elu<<<((size_t)N * K + 255) / 256, 256, 0, stream>>>(agg, bias, g, be, stats,
                                                                  out, N, K);
}

extern "C" void kernel_launch(void* const* d_in, const int* in_sizes, int n_in,
                              void* d_out, int out_size, void* d_ws, size_t ws_size,
                              hipStream_t stream) {
    const float* x   = (const float*)d_in[0];
    const int*   ei  = (const int*)d_in[1];
    const float* W1  = (const float*)d_in[2];
    const float* as1 = (const float*)d_in[3];
    const float* ad1 = (const float*)d_in[4];
    const float* b1  = (const float*)d_in[5];
    const float* g1  = (const float*)d_in[6];
    const float* be1 = (const float*)d_in[7];
    const float* W2  = (const float*)d_in[8];
    const float* as2 = (const float*)d_in[9];
    const float* ad2 = (const float*)d_in[10];
    const float* b2  = (const float*)d_in[11];
    const float* g2  = (const float*)d_in[12];
    const float* be2 = (const float*)d_in[13];
    const float* W3  = (const float*)d_in[14];
    const float* as3 = (const float*)d_in[15];
    const float* ad3 = (const float*)d_in[16];
    const float* b3  = (const float*)d_in[17];

    const int Fin = 128;
    const int N = in_sizes[0] / Fin;
    const int E = in_sizes[1] / 2;
    const int Etot = E + N;
    const int* srcp = ei;
    const int* dstp = ei + E;

    // workspace layout (fp32 elements)
    float* ws    = (float*)d_ws;
    float* bufA  = ws;                            // N*256  (h buffer)
    float* bufB  = bufA + (size_t)N * 256;        // N*256  (feat ping-pong)
    float* agg   = bufB + (size_t)N * 256;        // N*256
    float* pbuf  = agg + (size_t)N * 256;         // Etot*4
    float* als   = pbuf + (size_t)Etot * 4;       // N*4
    float* ald   = als + (size_t)N * 4;           // N*4
    float* ssum  = ald + (size_t)N * 4;           // N*4
    int*   menc  = (int*)(ssum + (size_t)N * 4);  // N*4
    float* stats = (float*)(menc + (size_t)N * 4);// 4*128

    // ---- layer 1: x[N,128] -> h1 -> agg -> bn/elu -> bufB[N,128] ----
    run_gat_layer(x, Fin, W1, as1, ad1, 32, srcp, dstp, E, N,
                  bufA, agg, pbuf, als, ald, ssum, menc, stream);
    run_bn_elu(agg, b1, g1, be1, stats, bufB, N, 128, stream);

    // ---- layer 2: bufB -> h2(bufA) -> agg -> bn/elu -> bufB ----
    run_gat_layer(bufB, 128, W2, as2, ad2, 32, srcp, dstp, E, N,
                  bufA, agg, pbuf, als, ald, ssum, menc, stream);
    run_bn_elu(agg, b2, g2, be2, stats, bufB, N, 128, stream);

    // ---- layer 3: bufB -> h3(bufA)[N,256] -> agg[N,4,64] -> head mean + b3 ----
    run_gat_layer(bufB, 128, W3, as3, ad3, 64, srcp, dstp, E, N,
                  bufA, agg, pbuf, als, ald, ssum, menc, stream);
    head_mean_bias<<<(N * 64 + 255) / 256, 256, 0, stream>>>(agg, b3, (float*)d_out, N);
}